// HumanLiker_41970420418117
// MI455X (gfx1250) — compile-verified
//
#include <hip/hip_runtime.h>
#include <hip/hip_bf16.h>
#include <stdint.h>

typedef unsigned int u32;
typedef unsigned long long u64;
typedef __attribute__((ext_vector_type(2))) float v2f;
typedef __attribute__((ext_vector_type(8))) float v8f;
typedef int v4i __attribute__((vector_size(4 * sizeof(int))));

#define BATCH 64
#define M_TOT 32736
#define PRE_K 1000
#define PAD_K 1024
#define POST_K 100
#define NW 64              // 16-bit mask words per row (64*16 = 1024 cols)
#define IMG_W 1536.0f
#define IMG_H 1024.0f
#define NMS_TH 0.6f
#define SCORE_TH 0.05f

#if defined(__has_builtin)
#  if __has_builtin(__builtin_amdgcn_global_load_async_to_lds_b128)
#    define HAVE_ASYNC_LDS 1
#  endif
#  if __has_builtin(__builtin_amdgcn_s_wait_asynccnt)
#    define HAVE_WAIT_ASYNC 1
#  endif
#endif

typedef __attribute__((address_space(1))) v4i GV4;   // global int4*
typedef __attribute__((address_space(3))) v4i LV4;   // LDS int4*

// ---------------------------------------------------------------- decode ----
__global__ void k_decode(const float* __restrict__ hm0, const float* __restrict__ hm1,
                         const float* __restrict__ hm2, const float* __restrict__ hm3,
                         const float* __restrict__ hm4,
                         const float* __restrict__ rg0, const float* __restrict__ rg1,
                         const float* __restrict__ rg2, const float* __restrict__ rg3,
                         const float* __restrict__ rg4,
                         const float* __restrict__ of0, const float* __restrict__ of1,
                         const float* __restrict__ of2, const float* __restrict__ of3,
                         const float* __restrict__ of4,
                         float* __restrict__ scores, float* __restrict__ boxes) {
  int m = blockIdx.x * 256 + threadIdx.x;
  int b = blockIdx.y;
  if (m >= M_TOT) return;

  const float *hm, *rg, *of;
  int p, W;
  float s;
  if (m < 24576)      { p = m;         W = 192; s = 8.f;   hm = hm0; rg = rg0; of = of0; }
  else if (m < 30720) { p = m - 24576; W = 96;  s = 16.f;  hm = hm1; rg = rg1; of = of1; }
  else if (m < 32256) { p = m - 30720; W = 48;  s = 32.f;  hm = hm2; rg = rg2; of = of2; }
  else if (m < 32640) { p = m - 32256; W = 24;  s = 64.f;  hm = hm3; rg = rg3; of = of3; }
  else                { p = m - 32640; W = 12;  s = 128.f; hm = hm4; rg = rg4; of = of4; }
  int H = (W * 2) / 3;           // all levels have H = (2/3)W
  int hw = H * W;
  int iy = p / W, ix = p - iy * W;

  float logit = hm[(size_t)b * hw + p];
  float sc = 1.0f / (1.0f + expf(-logit));
  sc = (sc > SCORE_TH) ? sc : 0.0f;

  const float* rb = rg + (size_t)b * 4 * hw + p;
  float d0 = fmaxf(rb[0], 0.f) * s;
  float d1 = fmaxf(rb[(size_t)hw], 0.f) * s;
  float d2 = fmaxf(rb[(size_t)2 * hw], 0.f) * s;
  float d3 = fmaxf(rb[(size_t)3 * hw], 0.f) * s;
  const float* ob = of + (size_t)b * 2 * hw + p;
  float cx = (float)ix * s + s * 0.5f + ob[0] * s;
  float cy = (float)iy * s + s * 0.5f + ob[(size_t)hw] * s;

  float4 box;
  box.x = fminf(fmaxf(cx - d0, 0.f), IMG_W);
  box.y = fminf(fmaxf(cy - d1, 0.f), IMG_H);
  box.z = fminf(fmaxf(cx + d2, 0.f), IMG_W);
  box.w = fminf(fmaxf(cy + d3, 0.f), IMG_H);

  scores[(size_t)b * M_TOT + m] = sc;
  ((float4*)boxes)[(size_t)b * M_TOT + m] = box;
}

// ------------------------------------------- exact top-1000 (radix + sort) --
__global__ void k_topk(const float* __restrict__ scores, const float* __restrict__ boxes,
                       float* __restrict__ topS, float* __restrict__ topBox) {
  __shared__ u32 hist[256];
  __shared__ u64 keys[PAD_K];
  __shared__ int tieIdx[256];
  __shared__ u32 sFixed;
  __shared__ int sRem;
  __shared__ u32 sGt, sTie;

  int b = blockIdx.x, tid = threadIdx.x;
  const float* sc = scores + (size_t)b * M_TOT;

  if (tid == 0) { sFixed = 0u; sRem = PRE_K; }
  __syncthreads();

  for (int kp = 3; kp >= 0; --kp) {
    if (tid < 256) hist[tid] = 0u;
    __syncthreads();
    u32 fHi = sFixed;
    for (int m = tid; m < M_TOT; m += 256) {
      u32 bits = __float_as_uint(sc[m]);
      bool ok = true;
      if (kp < 3) ok = ((bits >> ((kp + 1) * 8)) == fHi);
      if (ok) atomicAdd(&hist[(bits >> (kp * 8)) & 255u], 1u);
    }
    __syncthreads();
    if (tid == 0) {
      u32 cum = 0; int rem = sRem;
      for (int bin = 255; bin >= 0; --bin) {
        cum += hist[bin];
        if ((int)cum >= rem) {
          sFixed = (fHi << 8) | (u32)bin;
          sRem = rem - (int)(cum - hist[bin]);
          break;
        }
      }
    }
    __syncthreads();
  }
  u32 T = sFixed;
  int need = sRem;

  for (int i = tid; i < PAD_K; i += 256) keys[i] = 0ull;
  if (tid == 0) { sGt = 0u; sTie = 0u; }
  __syncthreads();

  for (int m = tid; m < M_TOT; m += 256) {
    u32 bits = __float_as_uint(sc[m]);
    if (bits > T) {
      u32 pos = atomicAdd(&sGt, 1u);
      keys[pos] = ((u64)bits << 32) | (u64)(0xFFFFFFFFu - (u32)m);
    } else if (bits == T) {
      u32 tp = atomicAdd(&sTie, 1u);
      if (tp < 256u) tieIdx[tp] = m;
    }
  }
  __syncthreads();
  int nTie = (int)min(sTie, 256u);
  u32 cGt = sGt;
  if (tid < nTie) {
    int my = tieIdx[tid];
    int rank = 0;
    for (int j = 0; j < nTie; ++j) rank += (tieIdx[j] < my);
    if (rank < need) keys[cGt + rank] = ((u64)T << 32) | (u64)(0xFFFFFFFFu - (u32)my);
  }
  __syncthreads();

  // bitonic sort, descending on (score_bits, ~index)
  for (int k = 2; k <= PAD_K; k <<= 1) {
    for (int j = k >> 1; j > 0; j >>= 1) {
      for (int i = tid; i < PAD_K; i += 256) {
        int ixj = i ^ j;
        if (ixj > i) {
          u64 a = keys[i], c = keys[ixj];
          if ((a < c) == ((i & k) == 0)) { keys[i] = c; keys[ixj] = a; }
        }
      }
      __syncthreads();
    }
  }

  for (int i = tid; i < PAD_K; i += 256) {
    u64 key = keys[i];
    u32 bits = (u32)(key >> 32);
    u32 idx = 0xFFFFFFFFu - (u32)(key & 0xFFFFFFFFull);
    float sv = 0.0f;
    float4 bx = make_float4(0.f, 0.f, 0.f, 0.f);
    if (idx < (u32)M_TOT) {
      sv = __uint_as_float(bits);
      bx = ((const float4*)boxes)[(size_t)b * M_TOT + idx];
    }
    topS[(size_t)b * PAD_K + i] = sv;
    ((float4*)topBox)[(size_t)b * PAD_K + i] = bx;
  }
}

// -------------------------------- IoU suppression masks (WMMA area outer-sum)
__global__ void k_iou(const float* __restrict__ topBox, u32* __restrict__ mask) {
  __shared__ float4 sbox[PAD_K];
  __shared__ float sarea[PAD_K];
  int tid = threadIdx.x;
  int itile = blockIdx.x;
  int b = blockIdx.y;

#ifdef HAVE_ASYNC_LDS
  {
    const float4* gsrc = (const float4*)topBox + (size_t)b * PAD_K;
    for (int k = tid; k < PAD_K; k += 256) {
      GV4* g = (GV4*)(unsigned long long)(uintptr_t)(gsrc + k);
      LV4* l = (LV4*)(u32)(uintptr_t)(&sbox[k]);
      __builtin_amdgcn_global_load_async_to_lds_b128(g, l, 0, 0);
    }
#  ifdef HAVE_WAIT_ASYNC
    __builtin_amdgcn_s_wait_asynccnt(0);
#  else
    asm volatile("s_wait_asynccnt 0x0" ::: "memory");
#  endif
    __syncthreads();
    for (int k = tid; k < PAD_K; k += 256) {
      float4 v = sbox[k];
      sarea[k] = (v.z - v.x) * (v.w - v.y);
    }
  }
#else
  for (int k = tid; k < PAD_K; k += 256) {
    float4 v = ((const float4*)topBox)[(size_t)b * PAD_K + k];
    sbox[k] = v;
    sarea[k] = (v.z - v.x) * (v.w - v.y);
  }
#endif
  __syncthreads();

  int lane = tid & 31;
  int wv = tid >> 5;
  int i0 = itile * 16;
  int mh = (lane >> 4) * 8;          // row half offset per C/D layout
  int n = lane & 15;

  // WMMA A (16x4 f32): col0 = rowArea, col1 = 1, cols 2..3 = 0
  v2f A;
  A.x = (lane < 16) ? sarea[i0 + lane] : 0.0f;   // K=0 (lanes 0-15) / K=2 (=0)
  A.y = (lane < 16) ? 1.0f : 0.0f;               // K=1 (lanes 0-15) / K=3 (=0)

  float rx1[8], ry1[8], rx2[8], ry2[8];
  for (int v = 0; v < 8; ++v) {
    float4 rb = sbox[i0 + v + mh];
    rx1[v] = rb.x; ry1[v] = rb.y; rx2[v] = rb.z; ry2[v] = rb.w;
  }

  u32* mrow = mask + (size_t)b * PAD_K * NW;
  for (int jt = wv; jt < NW; jt += 8) {
    int j0 = jt * 16;
    // WMMA B (4x16 f32): row0 = 1, row1 = colArea, rows 2..3 = 0
    v2f Bv;
    Bv.x = (lane < 16) ? 1.0f : 0.0f;                 // K=0 / K=2
    Bv.y = (lane < 16) ? sarea[j0 + lane] : 0.0f;     // K=1 / K=3
    v8f acc = {};
    acc = __builtin_amdgcn_wmma_f32_16x16x4_f32(
        false, A, false, Bv, (short)0, acc, false, false);
    // acc[v] = area_row[i0 + v + mh] + area_col[j0 + n]

    int gj = j0 + n;
    float4 cb = sbox[gj];
    for (int v = 0; v < 8; ++v) {
      int gi = i0 + v + mh;
      float lx = fmaxf(rx1[v], cb.x), ly = fmaxf(ry1[v], cb.y);
      float hx = fminf(rx2[v], cb.z), hy = fminf(ry2[v], cb.w);
      float iw = fmaxf(hx - lx, 0.f), ih = fmaxf(hy - ly, 0.f);
      float inter = iw * ih;
      float denom = acc[v] - inter + 1e-6f;   // area_i + area_j - inter + eps  (> 0)
      bool pred = (inter > NMS_TH * denom) && (gj > gi);   // iou > th, division-free
      u32 bal = __builtin_amdgcn_ballot_w32(pred);
      // single predicated store: lane 0 -> row i0+v (low half), lane 16 -> row i0+v+8
      u32 val = (lane == 0) ? (bal & 0xFFFFu) : (bal >> 16);
      int row = i0 + v + ((lane == 16) ? 8 : 0);
      if ((lane & 15) == 0)
        mrow[(size_t)row * NW + jt] = val;
    }
  }
}

// ----------------------------------- sequential NMS + final top-100 output --
__global__ void k_nms(const float* __restrict__ topS, const float* __restrict__ topBox,
                      const u32* __restrict__ mask, float* __restrict__ out) {
  int b = blockIdx.x;
  int lane = threadIdx.x;
  const u32* mrow = mask + (size_t)b * PAD_K * NW;

  u32 keepA = 0xFFFFu;   // words [0..31]   : keep bits for rows 16w..16w+15
  u32 keepB = 0xFFFFu;   // words [32..63]
  u32 rA = mrow[lane], rB = mrow[lane + 32];

  for (int i = 0; i < PRE_K; ++i) {
    u32 nA = 0u, nB = 0u;
    if (i + 1 < PRE_K) {
      const u32* nx = mrow + (size_t)(i + 1) * NW;
      nA = nx[lane]; nB = nx[lane + 32];
    }
    if (i + 8 < PRE_K)
      __builtin_prefetch((const void*)(mrow + (size_t)(i + 8) * NW + lane), 0, 1);
    int wi = i >> 4;
    u32 wa = __shfl(keepA, wi & 31, 32);
    u32 wb = __shfl(keepB, wi & 31, 32);
    u32 word = (wi < 32) ? wa : wb;
    if ((word >> (i & 15)) & 1u) { keepA &= ~rA; keepB &= ~rB; }
    rA = nA; rB = nB;
  }

  // stable compaction: kept positives in order, then everything else in order
  __shared__ int sel[POST_K];
  int cnt = 0;   // uniform across lanes
  for (int pass = 0; pass < 2; ++pass) {
    for (int base = 0; base < PRE_K; base += 32) {
      int k = base + lane;
      bool valid = k < PRE_K;
      int wi = k >> 4;
      u32 wa = __shfl(keepA, wi & 31, 32);
      u32 wb = __shfl(keepB, wi & 31, 32);
      u32 word = (wi < 32) ? wa : wb;
      bool kept = valid && (((word >> (k & 15)) & 1u) != 0u);
      float s = valid ? topS[(size_t)b * PAD_K + k] : 0.0f;
      bool pos = kept && (s > 0.0f);
      bool flag = valid && ((pass == 0) ? pos : !pos);
      u32 bal = __builtin_amdgcn_ballot_w32(flag);
      int pre = __popc(bal & ((1u << lane) - 1u));
      if (flag && (cnt + pre) < POST_K)
        sel[cnt + pre] = k | (pass == 0 ? 0x10000 : 0);
      cnt += __popc(bal);
      if (cnt >= POST_K) break;
    }
    if (cnt >= POST_K) break;
  }
  __syncthreads();

  for (int r = lane; r < POST_K; r += 32) {
    int e = sel[r];
    int k = e & 0xFFFF;
    bool pos = (e >> 16) != 0;
    float4 bx = ((const float4*)topBox)[(size_t)b * PAD_K + k];
    float s = pos ? topS[(size_t)b * PAD_K + k] : 0.0f;
    float* o = out + ((size_t)b * POST_K + r) * 5;
    o[0] = bx.x; o[1] = bx.y; o[2] = bx.z; o[3] = bx.w; o[4] = s;
  }
}

// ---------------------------------------------------------------------------
extern "C" void kernel_launch(void* const* d_in, const int* in_sizes, int n_in,
                              void* d_out, int out_size, void* d_ws, size_t ws_size,
                              hipStream_t stream) {
  const float* hm[5]; const float* rg[5]; const float* of[5];
  for (int i = 0; i < 5; ++i) {
    hm[i] = (const float*)d_in[i];
    rg[i] = (const float*)d_in[5 + i];
    of[i] = (const float*)d_in[10 + i];
  }
  char* ws = (char*)d_ws;
  size_t o_scores = 0;
  size_t o_boxes  = o_scores + (size_t)BATCH * M_TOT * 4;        //  8.4 MB
  size_t o_topS   = o_boxes  + (size_t)BATCH * M_TOT * 16;       // 33.5 MB
  size_t o_topBox = o_topS   + (size_t)BATCH * PAD_K * 4;
  size_t o_mask   = o_topBox + (size_t)BATCH * PAD_K * 16;
  float* scores = (float*)(ws + o_scores);
  float* boxes  = (float*)(ws + o_boxes);
  float* topS   = (float*)(ws + o_topS);
  float* topBox = (float*)(ws + o_topBox);
  u32*   mask   = (u32*)(ws + o_mask);

  dim3 gDec((M_TOT + 255) / 256, BATCH);
  k_decode<<<gDec, 256, 0, stream>>>(hm[0], hm[1], hm[2], hm[3], hm[4],
                                     rg[0], rg[1], rg[2], rg[3], rg[4],
                                     of[0], of[1], of[2], of[3], of[4],
                                     scores, boxes);
  k_topk<<<BATCH, 256, 0, stream>>>(scores, boxes, topS, topBox);
  dim3 gIou(PAD_K / 16, BATCH);
  k_iou<<<gIou, 256, 0, stream>>>(topBox, mask);
  k_nms<<<BATCH, 32, 0, stream>>>(topS, topBox, mask, (float*)d_out);
}